// StiffnessMatrix_39926015984151
// MI455X (gfx1250) — compile-verified
//
#include <hip/hip_runtime.h>

// ---------------------------------------------------------------------------
// Kernel 1: zero the 7500x7500 output with 128-bit stores (global_store_b128).
// This is the dominant cost: 225 MB @ 23.3 TB/s ~= 10 us floor.
// ---------------------------------------------------------------------------
__global__ void StiffnessZero_kernel(float* __restrict__ out, long long n) {
    long long n4  = n >> 2;
    long long gid = (long long)blockIdx.x * blockDim.x + threadIdx.x;
    float4* __restrict__ o4 = reinterpret_cast<float4*>(out);
    if (gid < n4) {
        o4[gid] = make_float4(0.0f, 0.0f, 0.0f, 0.0f);
    }
    if (gid == 0) {
        for (long long i = (n4 << 2); i < n; ++i) out[i] = 0.0f;  // tail (none here)
    }
}

// ---------------------------------------------------------------------------
// Kernel 2: one thread per (edge, entry) pair; 10000*36 = 360k threads.
// Closed-form 6x6 beam stiffness entry + hardware float atomic scatter-add.
//
// Structure of the reference 6x6 (indices j in 0..5, jm = j%3, sigma = +1 if
// j<3 else -1, q(0)=sin, q(1)=cos, p(0)=cos, p(1)=sin, mu(0)=+1, mu(1)=-1):
//   rot[r][c] = sig_r*sig_c * 12*(q_r*q_c)          if rm<2 && cm<2
//             = -sig_c * (6L*q_c)                   if rm==2, cm<2
//             = -sig_r * (6L*q_r)                   if cm==2, rm<2
//             = (r==c ? 4 : 2) * L^2                if rm==2 && cm==2
//   lin[r][c] = sig_r*sig_c*mu_r*mu_c * (p_r*p_c)   if rm<2 && cm<2, else 0
//   K[r][c]   = rot*krot + lin*klin
// Multiplication association mirrors the reference so float32 results match.
// ---------------------------------------------------------------------------
__global__ void StiffnessAssemble_kernel(const float* __restrict__ coords,
                                         const int*   __restrict__ srcIdx,
                                         const int*   __restrict__ dstIdx,
                                         const float* __restrict__ Emod,
                                         const float* __restrict__ Avec,
                                         float* __restrict__ out,
                                         int n_edges, int dim) {
    int t = blockIdx.x * blockDim.x + threadIdx.x;
    int total = n_edges * 36;
    if (t >= total) return;
    int e = t / 36;
    int i = t - e * 36;

    // gfx1250 global_prefetch_b8 on the edge-index stream (speculative RT).
    if (e + 64 < n_edges) {
        __builtin_prefetch(&srcIdx[e + 64], 0, 0);
        __builtin_prefetch(&dstIdx[e + 64], 0, 0);
    }

    int s = srcIdx[e];
    int d = dstIdx[e];

    float ux = coords[2 * s + 0], uy = coords[2 * s + 1];
    float vx = coords[2 * d + 0], vy = coords[2 * d + 1];
    float dx = ux - vx;
    float dy = uy - vy;
    float L  = sqrtf(dx * dx + dy * dy);

    float cs = dx / L;          // cos =  delta_x / L
    float sn = -dy / L;         // sin = -delta_y / L   (reference convention)

    float Ae = Avec[e], Ee = Emod[e];
    float Ie   = (Ae * Ae) / 12.0f;        // square beam shape factor
    float krot = (Ee * Ie) / ((L * L) * L);
    float klin = (Ee * Ae) / L;

    int r  = i / 6;
    int c  = i - r * 6;
    int rm = (r >= 3) ? (r - 3) : r;
    int cm = (c >= 3) ? (c - 3) : c;

    float sig_r = (r < 3) ? 1.0f : -1.0f;
    float sig_c = (c < 3) ? 1.0f : -1.0f;
    float q_r   = (rm == 0) ? sn : cs;
    float q_c   = (cm == 0) ? sn : cs;
    float p_r   = (rm == 0) ? cs : sn;
    float p_c   = (cm == 0) ? cs : sn;
    float mu_r  = (rm == 0) ? 1.0f : -1.0f;
    float mu_c  = (cm == 0) ? 1.0f : -1.0f;

    float L2 = L * L;
    float hL = 6.0f * L;

    bool r2 = (rm == 2), c2 = (cm == 2);
    float rot, lin;
    if (r2 && c2) {
        rot = ((r == c) ? 4.0f : 2.0f) * L2;
        lin = 0.0f;
    } else if (r2) {
        rot = -sig_c * (hL * q_c);
        lin = 0.0f;
    } else if (c2) {
        rot = -sig_r * (hL * q_r);
        lin = 0.0f;
    } else {
        rot = (sig_r * sig_c) * (12.0f * (q_r * q_c));
        lin = ((sig_r * sig_c) * (mu_r * mu_c)) * (p_r * p_c);
    }

    float val = rot * krot + lin * klin;

    int grow = (r < 3) ? (3 * s + r) : (3 * d + (r - 3));
    int gcol = (c < 3) ? (3 * s + c) : (3 * d + (c - 3));

    // hardware global_atomic_add_f32 (bypasses CAS-loop fallback)
    unsafeAtomicAdd(&out[(size_t)grow * (size_t)dim + (size_t)gcol], val);
}

extern "C" void kernel_launch(void* const* d_in, const int* in_sizes, int n_in,
                              void* d_out, int out_size, void* d_ws, size_t ws_size,
                              hipStream_t stream) {
    const float* coords = (const float*)d_in[0];
    const int*   src    = (const int*)d_in[1];
    const int*   dst    = (const int*)d_in[2];
    const float* E_mod  = (const float*)d_in[3];
    const float* A      = (const float*)d_in[4];
    float* out = (float*)d_out;

    const int n_edges = in_sizes[1];
    const int n_nodes = in_sizes[0] / 2;
    const int dim     = n_nodes * 3;

    // 1) zero-fill output (B128 stores)
    long long n  = (long long)out_size;
    long long n4 = n >> 2;
    int zblocks = (int)((n4 + 255) / 256);
    if (zblocks < 1) zblocks = 1;
    StiffnessZero_kernel<<<zblocks, 256, 0, stream>>>(out, n);

    // 2) per-entry compute + atomic scatter (stream-ordered after the fill)
    int total   = n_edges * 36;
    int ablocks = (total + 255) / 256;
    StiffnessAssemble_kernel<<<ablocks, 256, 0, stream>>>(coords, src, dst,
                                                          E_mod, A, out,
                                                          n_edges, dim);
}